// FrequencySemanticCipher_7370163880080
// MI455X (gfx1250) — compile-verified
//
#include <hip/hip_runtime.h>
#include <stdint.h>

// ---------------------------------------------------------------------------
// FrequencySemanticCipher for MI455X (gfx1250).
// enc = image + idwt(normalized_threefry_noise); then per-batch minmax norm.
// Pass A: noise maxabs per subband (compute-only + atomics)
// Pass B: enc = img + idwt(P), per-batch min/max  (100MB r + 100MB w)
// Pass C: normalize, streamed through LDS via TDM tensor_load_to_lds
// ---------------------------------------------------------------------------

#define TPB    256
#define HALF   3145728      // (32*3*256*256)/2 threefry pairs per subband
#define EPSF   1e-8f
#define CHUNK  8192         // floats per TDM tile (32 KB)
#define NCHUNK 3072         // 25165824 / 8192
#define CHUNKS_PER_BATCH 96 // 786432 / 8192

struct Keys8 { uint32_t k[8]; }; // ll0,ll1, lh0,lh1, hl0,hl1, hh0,hh1

// --- threefry2x32, matches JAX (20 rounds, key injection every 4) ----------
__host__ __device__ inline void tf2x32(uint32_t k0, uint32_t k1,
                                       uint32_t x0, uint32_t x1,
                                       uint32_t& o0, uint32_t& o1) {
  uint32_t ks2 = k0 ^ k1 ^ 0x1BD11BDAu;
  x0 += k0; x1 += k1;
#define TFR(r) { x0 += x1; x1 = (x1 << (r)) | (x1 >> (32 - (r))); x1 ^= x0; }
  TFR(13) TFR(15) TFR(26) TFR(6)
  x0 += k1;  x1 += ks2 + 1u;
  TFR(17) TFR(29) TFR(16) TFR(24)
  x0 += ks2; x1 += k0 + 2u;
  TFR(13) TFR(15) TFR(26) TFR(6)
  x0 += k0;  x1 += k1 + 3u;
  TFR(17) TFR(29) TFR(16) TFR(24)
  x0 += k1;  x1 += ks2 + 4u;
  TFR(13) TFR(15) TFR(26) TFR(6)
  x0 += ks2; x1 += k0 + 5u;
#undef TFR
  o0 = x0; o1 = x1;
}

// --- erfinv (Giles single-precision poly, same as XLA ErfInv32) ------------
__device__ inline float erfinv_f(float x) {
  float w = -log1pf(-x * x);
  float p;
  if (w < 5.0f) {
    w -= 2.5f;
    p = 2.81022636e-08f;
    p = fmaf(p, w, 3.43273939e-07f);
    p = fmaf(p, w, -3.5233877e-06f);
    p = fmaf(p, w, -4.39150654e-06f);
    p = fmaf(p, w, 0.00021858087f);
    p = fmaf(p, w, -0.00125372503f);
    p = fmaf(p, w, -0.00417768164f);
    p = fmaf(p, w, 0.246640727f);
    p = fmaf(p, w, 1.50140941f);
  } else {
    w = sqrtf(w) - 3.0f;
    p = -0.000200214257f;
    p = fmaf(p, w, 0.000100950558f);
    p = fmaf(p, w, 0.00134934322f);
    p = fmaf(p, w, -0.00367342844f);
    p = fmaf(p, w, 0.00573950773f);
    p = fmaf(p, w, -0.0076224613f);
    p = fmaf(p, w, 0.00943887047f);
    p = fmaf(p, w, 1.00167406f);
    p = fmaf(p, w, 2.83297682f);
  }
  return p * x;
}

// bits -> uniform(-0.99999994, 1) -> sqrt(2)*erfinv  (JAX random.normal)
__device__ inline float bits_to_normal(uint32_t bits) {
  float f01 = __uint_as_float((bits >> 9) | 0x3F800000u) - 1.0f; // [0,1)
  const float lo = -0.99999994f;
  float u = fmaxf(fmaf(f01, 1.0f - lo, lo), lo);
  return 1.4142135623730951f * erfinv_f(u);
}

// monotone float<->uint map for atomic min/max
__device__ inline uint32_t fmap(float f) {
  uint32_t u = __float_as_uint(f);
  return (u & 0x80000000u) ? ~u : (u | 0x80000000u);
}
__device__ inline float funmap(uint32_t m) {
  return (m & 0x80000000u) ? __uint_as_float(m ^ 0x80000000u)
                           : __uint_as_float(~m);
}

// ---------------------------------------------------------------------------
// ws layout (uint32): [0..3] subband maxabs bits, [4..35] batch max (mapped),
//                     [36..67] batch min (mapped)
// ---------------------------------------------------------------------------
__global__ void init_ws_kernel(uint32_t* __restrict__ ws) {
  int t = threadIdx.x;
  if (t < 36)           ws[t] = 0u;
  else if (t < 68)      ws[t] = 0xFFFFFFFFu;
}

// --- Pass A: max|n| per subband --------------------------------------------
__global__ void noise_maxabs_kernel(Keys8 keys, uint32_t* __restrict__ ws_max) {
  int sub = blockIdx.y;
  int i   = blockIdx.x * TPB + threadIdx.x;
  uint32_t o0, o1;
  tf2x32(keys.k[2 * sub], keys.k[2 * sub + 1],
         (uint32_t)i, (uint32_t)(i + HALF), o0, o1);
  float m = fmaxf(fabsf(bits_to_normal(o0)), fabsf(bits_to_normal(o1)));
  for (int off = 16; off > 0; off >>= 1)
    m = fmaxf(m, __shfl_xor(m, off));
  __shared__ float sred[TPB / 32];
  if ((threadIdx.x & 31) == 0) sred[threadIdx.x >> 5] = m;
  __syncthreads();
  if (threadIdx.x == 0) {
    float mm = sred[0];
#pragma unroll
    for (int q = 1; q < TPB / 32; ++q) mm = fmaxf(mm, sred[q]);
    atomicMax(&ws_max[sub], __float_as_uint(mm)); // mm >= 0: bit order == float order
  }
}

// --- Pass B: enc = img + idwt(P), track per-batch min/max ------------------
__global__ void encode_kernel(const float* __restrict__ img,
                              float* __restrict__ out, Keys8 keys,
                              const uint32_t* __restrict__ ws_maxabs,
                              uint32_t* __restrict__ ws_bmax,
                              uint32_t* __restrict__ ws_bmin) {
  int t = blockIdx.x * TPB + threadIdx.x;
  float p[2][4];
#pragma unroll
  for (int s = 0; s < 4; ++s) {
    uint32_t o0, o1;
    tf2x32(keys.k[2 * s], keys.k[2 * s + 1],
           (uint32_t)t, (uint32_t)(t + HALF), o0, o1);
    float inv = 1.0f / (__uint_as_float(ws_maxabs[s]) + EPSF);
    p[0][s] = bits_to_normal(o0) * inv;
    p[1][s] = bits_to_normal(o1) * inv;
  }
  float mn[2], mx[2];
#pragma unroll
  for (int h = 0; h < 2; ++h) {
    int idx = t + h * HALF;
    int j  = idx & 255;
    int i2 = (idx >> 8) & 255;
    int bc = idx >> 16;                       // b*3 + c
    long base = ((long)bc * 512 + 2 * i2) * 512 + 2 * j;
    float LL = p[h][0], LH = p[h][1], HL = p[h][2], HH = p[h][3];
    float da = (LL + LH + HL + HH) * 0.5f;
    float db = (LL + LH - HL - HH) * 0.5f;
    float dc = (LL - LH + HL - HH) * 0.5f;
    float dd = (LL - LH - HL + HH) * 0.5f;
    float2 r0 = *(const float2*)(img + base);
    float2 r1 = *(const float2*)(img + base + 512);
    float ea = r0.x + da, eb = r0.y + db;
    float ec = r1.x + dc, ed = r1.y + dd;
    *(float2*)(out + base)       = make_float2(ea, eb);
    *(float2*)(out + base + 512) = make_float2(ec, ed);
    mn[h] = fminf(fminf(ea, eb), fminf(ec, ed));
    mx[h] = fmaxf(fmaxf(ea, eb), fmaxf(ec, ed));
  }
  // block reduce (all threads of a block share the same batch pair b, b+16)
  for (int off = 16; off > 0; off >>= 1) {
    mn[0] = fminf(mn[0], __shfl_xor(mn[0], off));
    mx[0] = fmaxf(mx[0], __shfl_xor(mx[0], off));
    mn[1] = fminf(mn[1], __shfl_xor(mn[1], off));
    mx[1] = fmaxf(mx[1], __shfl_xor(mx[1], off));
  }
  __shared__ float sred[4][TPB / 32];
  int lane = threadIdx.x & 31, w = threadIdx.x >> 5;
  if (lane == 0) {
    sred[0][w] = mn[0]; sred[1][w] = mx[0];
    sred[2][w] = mn[1]; sred[3][w] = mx[1];
  }
  __syncthreads();
  if (threadIdx.x == 0) {
    float a0 = sred[0][0], a1 = sred[1][0], a2 = sred[2][0], a3 = sred[3][0];
#pragma unroll
    for (int q = 1; q < TPB / 32; ++q) {
      a0 = fminf(a0, sred[0][q]); a1 = fmaxf(a1, sred[1][q]);
      a2 = fminf(a2, sred[2][q]); a3 = fmaxf(a3, sred[3][q]);
    }
    int b0 = ((blockIdx.x * TPB) >> 16) / 3;   // batch of h=0; h=1 is b0+16
    atomicMin(&ws_bmin[b0],      fmap(a0));
    atomicMax(&ws_bmax[b0],      fmap(a1));
    atomicMin(&ws_bmin[b0 + 16], fmap(a2));
    atomicMax(&ws_bmax[b0 + 16], fmap(a3));
  }
}

// --- Pass C: per-batch normalize, streamed via TDM tensor_load_to_lds ------
typedef unsigned int v4u __attribute__((ext_vector_type(4)));
typedef int          v8i __attribute__((ext_vector_type(8)));
typedef int          v4i __attribute__((ext_vector_type(4)));

__global__ void normalize_kernel(float* __restrict__ out,
                                 const uint32_t* __restrict__ ws_bmax,
                                 const uint32_t* __restrict__ ws_bmin) {
  __shared__ float tile[CHUNK];
  long base = (long)blockIdx.x * CHUNK;
  int  b    = blockIdx.x / CHUNKS_PER_BATCH;
  float mx = funmap(ws_bmax[b]);
  float mn = funmap(ws_bmin[b]);
  float rs = 1.0f / fmaxf(mx - mn, EPSF);

#if defined(__HIP_DEVICE_COMPILE__)
  if (threadIdx.x < 32) {  // wave 0 issues one 32KB TDM load into LDS
    uint64_t ga = (uint64_t)(uintptr_t)(out + base);
    uint32_t la = (uint32_t)(uintptr_t)(&tile[0]); // low 32 bits = LDS offset
    // D# group0: count=1 | lds_addr | global_addr[56:0] | type=2
    v4u g0 = { 1u, la, (uint32_t)ga,
               (uint32_t)((ga >> 32) & 0x01FFFFFFu) | 0x80000000u };
    // D# group1: data_size=4B, tensor_dim0=tile_dim0=CHUNK, dim1=1, strides=CHUNK
    v8i g1 = { 0x20000,
               (int)((CHUNK & 0xFFFF) << 16),               // tensor_dim0 lo16
               (int)(((CHUNK >> 16) & 0xFFFF) | (1 << 16)), // dim0 hi | tensor_dim1 lo
               (int)((CHUNK & 0xFFFF) << 16),               // tile_dim0
               1,                                           // tile_dim1=1, tile_dim2=0
               (int)CHUNK,                                  // tensor_dim0_stride lo32
               (int)((CHUNK & 0xFFFF) << 16),               // d0 stride hi | d1 stride lo
               0 };
    v4i g2 = { 0, 0, 0, 0 };
    v4i g3 = { 0, 0, 0, 0 };
    v8i g4 = { 0, 0, 0, 0, 0, 0, 0, 0 };
    // 6-arg form on this toolchain: (g0, g1, g2, g3, g4, cpol)
    __builtin_amdgcn_tensor_load_to_lds(g0, g1, g2, g3, g4, 0);
    __builtin_amdgcn_s_wait_tensorcnt(0);
  }
#endif
  __syncthreads();

#pragma unroll
  for (int q = 0; q < CHUNK / (TPB * 4); ++q) {
    int off = (q * TPB + threadIdx.x) * 4;
    float4 v = *(const float4*)&tile[off];
    v.x = (v.x - mn) * rs;
    v.y = (v.y - mn) * rs;
    v.z = (v.z - mn) * rs;
    v.w = (v.w - mn) * rs;
    *(float4*)(out + base + off) = v;
  }
}

// ---------------------------------------------------------------------------
extern "C" void kernel_launch(void* const* d_in, const int* in_sizes, int n_in,
                              void* d_out, int out_size, void* d_ws,
                              size_t ws_size, hipStream_t stream) {
  const float* img = (const float*)d_in[0];
  float* out       = (float*)d_out;
  uint32_t* ws     = (uint32_t*)d_ws;

  // jax.random.split(key(123), 4) on host: threefry over counters (i, i+4)
  uint32_t a[4], b[4];
  for (uint32_t i = 0; i < 4; ++i)
    tf2x32(0u, 123u, i, i + 4u, a[i], b[i]);
  Keys8 keys;
  keys.k[0] = a[0]; keys.k[1] = a[1];   // k_ll
  keys.k[2] = a[2]; keys.k[3] = a[3];   // k_lh
  keys.k[4] = b[0]; keys.k[5] = b[1];   // k_hl
  keys.k[6] = b[2]; keys.k[7] = b[3];   // k_hh

  init_ws_kernel<<<1, 128, 0, stream>>>(ws);
  dim3 gA(HALF / TPB, 4);
  noise_maxabs_kernel<<<gA, TPB, 0, stream>>>(keys, ws);
  encode_kernel<<<HALF / TPB, TPB, 0, stream>>>(img, out, keys,
                                                ws, ws + 4, ws + 36);
  normalize_kernel<<<NCHUNK, TPB, 0, stream>>>(out, ws + 4, ws + 36);
}